// RFA_72464688218293
// MI455X (gfx1250) — compile-verified
//
#include <hip/hip_runtime.h>

typedef __attribute__((ext_vector_type(2))) float v2f;
typedef __attribute__((ext_vector_type(8))) float v8f;

#define D 32
#define KN 16

// ---------------- kernel 1: zero the tnet max accumulator ----------------
__global__ void k_init(unsigned* hmax, int n) {
    int i = blockIdx.x * blockDim.x + threadIdx.x;
    if (i < n) hmax[i] = 0u;
}

// ---------------- kernel 2: tnet MLP (3->16->64) + global max over N -----
__global__ void k_tnet(const float* __restrict__ xyz,
                       const float* __restrict__ tW1, const float* __restrict__ ts1, const float* __restrict__ tb1,
                       const float* __restrict__ tW2, const float* __restrict__ ts2, const float* __restrict__ tb2,
                       unsigned* __restrict__ hmax, int N, int BN) {
    __shared__ float lW1[48], ls1[16], lb1[16], lW2[16 * 64], ls2[64], lb2[64];
    int tid = threadIdx.x;
    for (int i = tid; i < 48; i += blockDim.x) lW1[i] = tW1[i];
    for (int i = tid; i < 16; i += blockDim.x) { ls1[i] = ts1[i]; lb1[i] = tb1[i]; }
    for (int i = tid; i < 1024; i += blockDim.x) lW2[i] = tW2[i];
    for (int i = tid; i < 64; i += blockDim.x) { ls2[i] = ts2[i]; lb2[i] = tb2[i]; }
    __syncthreads();

    int p = blockIdx.x * blockDim.x + tid;
    if (p >= BN) return;
    int b = p / N;

    float x0 = xyz[p * 3 + 0], x1 = xyz[p * 3 + 1], x2 = xyz[p * 3 + 2];
    float h1[16];
#pragma unroll
    for (int o = 0; o < 16; ++o) {
        float a = x0 * lW1[0 * 16 + o] + x1 * lW1[1 * 16 + o] + x2 * lW1[2 * 16 + o];
        h1[o] = fmaxf(0.f, a * ls1[o] + lb1[o]);
    }
    float h2[64];
#pragma unroll
    for (int o = 0; o < 64; ++o) {
        float a = 0.f;
#pragma unroll
        for (int i = 0; i < 16; ++i) a += h1[i] * lW2[i * 64 + o];
        h2[o] = fmaxf(0.f, a * ls2[o] + lb2[o]);
    }
    // wave32 butterfly max per channel
#pragma unroll
    for (int o = 0; o < 64; ++o) {
        float v = h2[o];
        v = fmaxf(v, __shfl_xor(v, 16, 32));
        v = fmaxf(v, __shfl_xor(v, 8, 32));
        v = fmaxf(v, __shfl_xor(v, 4, 32));
        v = fmaxf(v, __shfl_xor(v, 2, 32));
        v = fmaxf(v, __shfl_xor(v, 1, 32));
        h2[o] = v;
    }
    int lane = tid & 31;
    // relu() outputs are >= 0 so uint-bit-compare == float-compare; init 0 is the relu floor
    atomicMax(&hmax[b * 64 + lane], __float_as_uint(h2[lane]));
    atomicMax(&hmax[b * 64 + 32 + lane], __float_as_uint(h2[32 + lane]));
}

// ---------------- kernel 3: h3, h4+eye -> t ; M = t @ W1 -----------------
__global__ void k_tmat(const unsigned* __restrict__ hmax,
                       const float* __restrict__ tW3, const float* __restrict__ ts3, const float* __restrict__ tb3,
                       const float* __restrict__ tW4, const float* __restrict__ tb4,
                       const float* __restrict__ W1, float* __restrict__ Mout, int B) {
    __shared__ float h3s[64];   // up to B=4
    __shared__ float tmat[36];
    int tid = threadIdx.x;
    if (tid < B * 16) {
        int b = tid >> 4, o = tid & 15;
        float a = 0.f;
        for (int i = 0; i < 64; ++i) a += __uint_as_float(hmax[b * 64 + i]) * tW3[i * 16 + o];
        h3s[tid] = fmaxf(0.f, a * ts3[o] + tb3[o]);
    }
    __syncthreads();
    if (tid < B * 9) {
        int b = tid / 9, j = tid % 9;
        float a = tb4[j];
        for (int i = 0; i < 16; ++i) a += h3s[b * 16 + i] * tW4[i * 9 + j];
        if (j == 0 || j == 4 || j == 8) a += 1.f;   // + eye(3)
        tmat[tid] = a;
    }
    __syncthreads();
    if (tid < B * 9) {
        int b = tid / 9, rc = tid % 9, r = rc / 3, c = rc % 3;
        float a = 0.f;
        for (int k = 0; k < 3; ++k) a += tmat[b * 9 + r * 3 + k] * W1[k * 3 + c];
        Mout[tid] = a;   // M = t @ W1  (no activation between them)
    }
}

// ---------------- kernel 4: f_xyz (padded to 4), featT, f_nei staging ----
__global__ void k_point(const float* __restrict__ feature, const float* __restrict__ xyz,
                        const float* __restrict__ Mmat,
                        const float* __restrict__ W2, const float* __restrict__ s2, const float* __restrict__ b2,
                        const float* __restrict__ s1, const float* __restrict__ b1,
                        float* __restrict__ f_xyz4, float* __restrict__ f_nei,
                        float* __restrict__ featT, int N, int BN) {
    __shared__ float lW2[D * D], ls2[D], lb2[D];
    int tid = threadIdx.x;
    for (int i = tid; i < D * D; i += blockDim.x) lW2[i] = W2[i];
    if (tid < D) { ls2[tid] = s2[tid]; lb2[tid] = b2[tid]; }
    __syncthreads();

    int p = blockIdx.x * blockDim.x + tid;
    if (p >= BN) return;
    int b = p / N, n = p % N;

    // f_xyz = relu((xyz @ M) * s1 + b1), stored as 4 floats/point with zero pad
    float x0 = xyz[p * 3 + 0], x1 = xyz[p * 3 + 1], x2 = xyz[p * 3 + 2];
    const float* M = Mmat + b * 9;
#pragma unroll
    for (int c = 0; c < 3; ++c) {
        float a = x0 * M[0 * 3 + c] + x1 * M[1 * 3 + c] + x2 * M[2 * 3 + c];
        f_xyz4[(size_t)p * 4 + c] = fmaxf(0.f, a * s1[c] + b1[c]);
    }
    f_xyz4[(size_t)p * 4 + 3] = 0.f;   // pad -> uniform v2f tail loads in k_main

    // transpose channel-major feature -> per-point contiguous (coalesced across lanes)
    float v[D];
    const float* fb = feature + (size_t)b * D * N + n;
#pragma unroll
    for (int i = 0; i < D; ++i) v[i] = fb[(size_t)i * N];
#pragma unroll
    for (int i = 0; i < D; ++i) featT[(size_t)p * D + i] = v[i];

    // f_nei = relu((feat @ W2) * s2 + b2)
#pragma unroll 4
    for (int o = 0; o < D; ++o) {
        float a = 0.f;
#pragma unroll
        for (int i = 0; i < D; ++i) a += v[i] * lW2[i * D + o];
        f_nei[(size_t)p * D + o] = fmaxf(0.f, a * ls2[o] + lb2[o]);
    }
}

// ---------------- kernel 5: WMMA gather-GEMM + Kmax + W4 stage -----------
// One wave per point: A = 16 neighbors x 36 (zero-padded feats), B = W3pad 36x32.
// 9 k-chunks x 2 n-tiles of v_wmma_f32_16x16x4_f32.
// W3 staged in LDS pair-interleaved so each B-fragment is one aligned ds_load_b64
// (pairs merge into ds_load_2addr_b64); every A-fragment is one global_load_b64.
__global__ void __launch_bounds__(256) k_main(
        const int* __restrict__ nidx,
        const float* __restrict__ f_nei, const float* __restrict__ f_xyz4,
        const float* __restrict__ featT,
        const float* __restrict__ W3, const float* __restrict__ s3, const float* __restrict__ b3,
        const float* __restrict__ W4, const float* __restrict__ s4, const float* __restrict__ b4,
        float* __restrict__ out, int N, int BN) {
    __shared__ float lW3t[18 * 64];         // [pi][col][2] pair-interleaved, row 35 = zero pad
    __shared__ float lW4[64 * 32];
    __shared__ float ls3[32], lb3[32], ls4[32], lb4[32];
    __shared__ float lf[8][64];             // [feat(32) | fcat(32)] per wave
    int tid = threadIdx.x;
    for (int i = tid; i < 36 * 32; i += blockDim.x) {
        int row = i >> 5, col = i & 31;
        float v = (row < 35) ? W3[row * 32 + col] : 0.f;
        lW3t[(row >> 1) * 64 + col * 2 + (row & 1)] = v;
    }
    for (int i = tid; i < 64 * 32; i += blockDim.x) lW4[i] = W4[i];
    if (tid < 32) { ls3[tid] = s3[tid]; lb3[tid] = b3[tid]; ls4[tid] = s4[tid]; lb4[tid] = b4[tid]; }
    __syncthreads();

    int w = tid >> 5, lane = tid & 31;
    int half = lane >> 4, lr = lane & 15;
    int p = blockIdx.x * 8 + w;
    bool valid = p < BN;
    int pc = valid ? p : BN - 1;            // clamp: keep wave-uniform EXEC for WMMA/barriers
    int b = pc / N, n = pc % N;

    // neighbor for A-matrix row M=lr (lanes l and l+16 share the row, split K halves)
    int idx = nidx[(size_t)pc * KN + lr];
    size_t q = (size_t)b * N + (size_t)idx;
    const float* fn = f_nei + q * D;
    const float* fx = f_xyz4 + q * 4;

    v8f acc0 = {}; v8f acc1 = {};
#pragma unroll
    for (int c = 0; c < 9; ++c) {
        int k  = 4 * c + 2 * half;          // this lane's K pair (rows k, k+1)
        int pi = 2 * c + half;              // pair-row index into lW3t
        v2f a = (c < 8) ? *(const v2f*)&fn[k]          // features 0..31
                        : *(const v2f*)&fx[2 * half];  // features 32..34 + zero pad
        v2f bb0 = *(const v2f*)&lW3t[pi * 64 + lr * 2];          // cols 0..15 tile
        v2f bb1 = *(const v2f*)&lW3t[pi * 64 + (16 + lr) * 2];   // cols 16..31 tile
        acc0 = __builtin_amdgcn_wmma_f32_16x16x4_f32(false, a, false, bb0, (short)0, acc0, false, false);
        acc1 = __builtin_amdgcn_wmma_f32_16x16x4_f32(false, a, false, bb1, (short)0, acc1, false, false);
    }

    // scale+bias+relu per element, then max over the 16 neighbor rows
    float sc0 = ls3[lr], bi0 = lb3[lr], sc1 = ls3[16 + lr], bi1 = lb3[16 + lr];
    float m0 = 0.f, m1 = 0.f;               // relu floor
#pragma unroll
    for (int r = 0; r < 8; ++r) {
        m0 = fmaxf(m0, fmaxf(0.f, acc0[r] * sc0 + bi0));
        m1 = fmaxf(m1, fmaxf(0.f, acc1[r] * sc1 + bi1));
    }
    m0 = fmaxf(m0, __shfl_xor(m0, 16, 32));
    m1 = fmaxf(m1, __shfl_xor(m1, 16, 32));

    // stash [feat | fcat] for the W4 stage
    lf[w][lane] = featT[(size_t)pc * D + lane];
    if (half == 0) { lf[w][32 + lr] = m0; lf[w][48 + lr] = m1; }
    __syncthreads();

    // out_c = relu((sum_i f_i * W4[i][c]) * s4 + b4), c = lane
    float a4 = 0.f;
#pragma unroll
    for (int i = 0; i < 64; ++i) a4 += lf[w][i] * lW4[i * 32 + lane];
    float o = fmaxf(0.f, a4 * ls4[lane] + lb4[lane]);
    if (valid) out[((size_t)b * 32 + lane) * N + n] = o;
}

// ---------------- launch ----------------
extern "C" void kernel_launch(void* const* d_in, const int* in_sizes, int n_in,
                              void* d_out, int out_size, void* d_ws, size_t ws_size,
                              hipStream_t stream) {
    const float* feature = (const float*)d_in[0];
    const float* xyz     = (const float*)d_in[1];
    const int*   nidx    = (const int*)d_in[2];
    const float* tW1 = (const float*)d_in[3],  *ts1 = (const float*)d_in[4],  *tb1 = (const float*)d_in[5];
    const float* tW2 = (const float*)d_in[6],  *ts2 = (const float*)d_in[7],  *tb2 = (const float*)d_in[8];
    const float* tW3 = (const float*)d_in[9],  *ts3 = (const float*)d_in[10], *tb3 = (const float*)d_in[11];
    const float* tW4 = (const float*)d_in[12], *tb4 = (const float*)d_in[13];
    const float* W1 = (const float*)d_in[14], *s1 = (const float*)d_in[15], *b1 = (const float*)d_in[16];
    const float* W2 = (const float*)d_in[17], *s2 = (const float*)d_in[18], *b2 = (const float*)d_in[19];
    const float* W3 = (const float*)d_in[20], *s3 = (const float*)d_in[21], *b3 = (const float*)d_in[22];
    const float* W4 = (const float*)d_in[23], *s4 = (const float*)d_in[24], *b4 = (const float*)d_in[25];
    float* out = (float*)d_out;

    int BN = in_sizes[1] / 3;               // B*N
    int N  = 65536;
    if (BN < N) N = BN;
    int B  = BN / N;

    // workspace layout
    char* ws = (char*)d_ws;
    unsigned* hmax = (unsigned*)(ws + 0);                 // B*64 uints
    float* Mmat    = (float*)(ws + 1024);                 // B*9
    float* fxyz4   = (float*)(ws + 2048);                 // BN*4 (zero-padded)
    size_t off = 2048 + (size_t)BN * 4 * sizeof(float);
    off = (off + 255) & ~(size_t)255;
    float* fnei = (float*)(ws + off);                     // BN*32
    off += (size_t)BN * D * sizeof(float);
    float* featT = (float*)(ws + off);                    // BN*32

    int blocks = (BN + 255) / 256;
    k_init<<<1, 256, 0, stream>>>(hmax, B * 64);
    k_tnet<<<blocks, 256, 0, stream>>>(xyz, tW1, ts1, tb1, tW2, ts2, tb2, hmax, N, BN);
    k_tmat<<<1, 64, 0, stream>>>(hmax, tW3, ts3, tb3, tW4, tb4, W1, Mmat, B);
    k_point<<<blocks, 256, 0, stream>>>(feature, xyz, Mmat, W2, s2, b2, s1, b1,
                                        fxyz4, fnei, featT, N, BN);
    k_main<<<(BN + 7) / 8, 256, 0, stream>>>(nidx, fnei, fxyz4, featT,
                                             W3, s3, b3, W4, s4, b4, out, N, BN);
}